// GraphProjection_2465311228494
// MI455X (gfx1250) — compile-verified
//
#include <hip/hip_runtime.h>
#include <hip/hip_bf16.h>
#include <math.h>

#ifndef __has_builtin
#define __has_builtin(x) 0
#endif

typedef unsigned int v4u __attribute__((ext_vector_type(4)));
typedef int   v8i __attribute__((ext_vector_type(8)));
typedef int   v4i __attribute__((ext_vector_type(4)));
typedef float v2f __attribute__((ext_vector_type(2)));
typedef float v8f __attribute__((ext_vector_type(8)));

#define PI_F 3.14159265358979323846f

// ---------------- LDS layout (bytes) ----------------
#define F2_ELEMS (256 * 196)            // feat2[0] : 256 ch x 14x14
#define F3_ELEMS (512 * 49)             // feat3[0] : 512 ch x 7x7
#define F2_OFF   0
#define F3_OFF   (F2_ELEMS * 4)         // 200704
#define M_OFF    (F3_OFF + F3_ELEMS*4)  // 301056 : combined 4x16 transform
#define PROJ_OFF (M_OFF + 4*16*4)       // 301312 : 128 points x 16 f32
#define OFFS_OFF (PROJ_OFF + 128*16*4)  // 309504 : 128 points x 12 int
#define LDS_BYTES (OFFS_OFF + 128*12*4) // 315648  (<= 320KB WGP LDS)

#define PTS_PER_BLOCK 128
#define THREADS 512

#if defined(__AMDGCN__) && __has_builtin(__builtin_amdgcn_tensor_load_to_lds)
#define HAVE_TDM 1
#else
#define HAVE_TDM 0
#endif

#if __has_builtin(__builtin_amdgcn_sqrtf)
#define FAST_SQRT(x) __builtin_amdgcn_sqrtf(x)
#else
#define FAST_SQRT(x) sqrtf(x)
#endif

#if HAVE_TDM
// Contiguous 1D global->LDS DMA of nelem dwords via the Tensor Data Mover.
// D# packing per CDNA5 ISA ch.8 (group0: count/lds_addr/global_addr/type,
// group1: data_size=4B, tensor_dim0=tile_dim0=nelem, stride=nelem).
// 6-arg builtin on this toolchain:
//   (uint32x4 g0, int32x8 g1, int32x4 g2, int32x4 g3, int32x8 g4, i32 cpol)
__device__ __forceinline__ void tdm_load_1d(const void* gptr, unsigned lds_byte,
                                            unsigned nelem) {
  unsigned long long ga = (unsigned long long)(size_t)gptr;
  v4u g0;
  g0[0] = 1u;                                          // count=1, user mode
  g0[1] = lds_byte;                                    // lds_addr [63:32]
  g0[2] = (unsigned)(ga & 0xFFFFFFFFu);                // global_addr[31:0]
  g0[3] = (unsigned)((ga >> 32) & 0x1FFFFFFu) | (2u << 30); // addr[56:32], type=2
  v8i g1;
  g1[0] = (int)(2u << 16);                             // data_size = 2 (4 bytes)
  g1[1] = (int)((nelem & 0xFFFFu) << 16);              // tensor_dim0[15:0] @ [63:48]
  g1[2] = (int)(((nelem >> 16) & 0xFFFFu) | (1u << 16)); // dim0 hi | tensor_dim1=1
  g1[3] = (int)((nelem & 0xFFFFu) << 16);              // tile_dim0 @ [127:112]
  g1[4] = 0;                                           // tile_dim1/2 unused
  g1[5] = (int)nelem;                                  // tensor_dim0_stride lo
  g1[6] = 0;
  g1[7] = 0;
  v4i z4 = {0, 0, 0, 0};
  v8i z8 = {0, 0, 0, 0, 0, 0, 0, 0};
  __builtin_amdgcn_tensor_load_to_lds(g0, g1, z4, z4, z8, 0);
}
#endif

__device__ __forceinline__ void wait_tdm_done() {
#if HAVE_TDM
#if __has_builtin(__builtin_amdgcn_s_wait_tensorcnt)
  __builtin_amdgcn_s_wait_tensorcnt(0);
#else
  asm volatile("s_wait_tensorcnt 0x0" ::: "memory");
#endif
#endif
}

__global__ __launch_bounds__(THREADS)
void graphproj_kernel(const float* __restrict__ inputs,
                      const float* __restrict__ cameras,
                      const float* __restrict__ feat0,
                      const float* __restrict__ feat1,
                      const float* __restrict__ feat2,
                      const float* __restrict__ feat3,
                      float* __restrict__ out, int N) {
  extern __shared__ char smem[];
  float* f2s  = (float*)(smem + F2_OFF);
  float* f3s  = (float*)(smem + F3_OFF);
  float* Ms   = (float*)(smem + M_OFF);    // [4][16] combined affine, cols 9..15 = 0
  float* proj = (float*)(smem + PROJ_OFF); // [128][16] projected coords (9 used)
  int*   offs = (int*)(smem + OFFS_OFF);   // [128][12] = [point][view*4+scale]

  const int tid  = threadIdx.x;
  const int lane = tid & 31;
  const int wave = tid >> 5;
  const int blockBase = blockIdx.x * PTS_PER_BLOCK;

  // ---- Phase 0: kick off TDM staging of feat2[0]/feat3[0] into LDS ----
#if HAVE_TDM
  if (wave == 0) {
    unsigned ldsbase = (unsigned)(size_t)smem; // LDS aperture: low 32 bits = byte offset
    tdm_load_1d(feat2, ldsbase + F2_OFF, F2_ELEMS);
    tdm_load_1d(feat3, ldsbase + F3_OFF, F3_ELEMS);
  }
#else
  for (int i = tid; i < F2_ELEMS; i += THREADS) f2s[i] = feat2[i];
  for (int i = tid; i < F3_ELEMS; i += THREADS) f3s[i] = feat3[i];
#endif

  // ---- Phase 1 (thread 0): camera mats, inv(c0^T), fold into 4x16 B matrix ----
  // Fast hardware trig (V_SIN/COS_F32): keeps the pre-barrier serial chain tiny.
  if (tid == 0) {
    float c[3][3][3], o[3][3];
#pragma unroll
    for (int v = 0; v < 3; ++v) {
      float th = cameras[v*5+0] * (PI_F/180.0f);
      float ph = cameras[v*5+1] * (PI_F/180.0f);
      float dist = cameras[v*5+3];
      float camy = dist * __sinf(ph);
      float lens = dist * __cosf(ph);
      float cth = __cosf(th), sth = __sinf(th);
      float camx = lens * cth;
      float camz = lens * sth;
      float Zax[3] = {camx, camy, camz};
      float Yax[3] = {-camy * cth, lens, -camy * sth};  // cos/sin(th+pi) = -cos/-sin
      float Xax[3] = {Yax[1]*Zax[2]-Yax[2]*Zax[1],
                      Yax[2]*Zax[0]-Yax[0]*Zax[2],
                      Yax[0]*Zax[1]-Yax[1]*Zax[0]};
      float nX = __frsqrt_rn(Xax[0]*Xax[0]+Xax[1]*Xax[1]+Xax[2]*Xax[2]);
      float nY = __frsqrt_rn(Yax[0]*Yax[0]+Yax[1]*Yax[1]+Yax[2]*Yax[2]);
      float nZ = __frsqrt_rn(Zax[0]*Zax[0]+Zax[1]*Zax[1]+Zax[2]*Zax[2]);
#pragma unroll
      for (int d = 0; d < 3; ++d) {
        c[v][0][d] = Xax[d]*nX; c[v][1][d] = Yax[d]*nY; c[v][2][d] = Zax[d]*nZ;
        o[v][d] = Zax[d];
      }
    }
    // Minv = inv(c0^T); T[i][j] = c0[j][i]
    float T[3][3];
#pragma unroll
    for (int i = 0; i < 3; ++i)
#pragma unroll
      for (int j = 0; j < 3; ++j) T[i][j] = c[0][j][i];
    float det = T[0][0]*(T[1][1]*T[2][2]-T[1][2]*T[2][1])
              - T[0][1]*(T[1][0]*T[2][2]-T[1][2]*T[2][0])
              + T[0][2]*(T[1][0]*T[2][1]-T[1][1]*T[2][0]);
    float id = 1.0f / det;
    float Minv[3][3];
    Minv[0][0] = (T[1][1]*T[2][2]-T[1][2]*T[2][1])*id;
    Minv[0][1] = (T[0][2]*T[2][1]-T[0][1]*T[2][2])*id;
    Minv[0][2] = (T[0][1]*T[1][2]-T[0][2]*T[1][1])*id;
    Minv[1][0] = (T[1][2]*T[2][0]-T[1][0]*T[2][2])*id;
    Minv[1][1] = (T[0][0]*T[2][2]-T[0][2]*T[2][0])*id;
    Minv[1][2] = (T[0][2]*T[1][0]-T[0][0]*T[1][2])*id;
    Minv[2][0] = (T[1][0]*T[2][1]-T[1][1]*T[2][0])*id;
    Minv[2][1] = (T[0][1]*T[2][0]-T[0][0]*T[2][1])*id;
    Minv[2][2] = (T[0][0]*T[1][1]-T[0][1]*T[1][0])*id;
#pragma unroll
    for (int i = 0; i < 64; ++i) Ms[i] = 0.0f;
    // pts_v[k] = sum_j p[j]*(Minv[j]·c_v[k]) + (o0-o_v)·c_v[k]  -> B[4][3v+k]
#pragma unroll
    for (int v = 0; v < 3; ++v)
#pragma unroll
      for (int k = 0; k < 3; ++k) {
        int col = 3*v + k;
#pragma unroll
        for (int j = 0; j < 3; ++j)
          Ms[j*16+col] = Minv[j][0]*c[v][k][0] + Minv[j][1]*c[v][k][1]
                       + Minv[j][2]*c[v][k][2];
        Ms[3*16+col] = (o[0][0]-o[v][0])*c[v][k][0] + (o[0][1]-o[v][1])*c[v][k][1]
                     + (o[0][2]-o[v][2])*c[v][k][2];
      }
  }
  __syncthreads();

  // ---- Phase 2: WMMA point transform, 16 points per wave (waves 0..7) ----
#if __has_builtin(__builtin_amdgcn_wmma_f32_16x16x4_f32)
  if (wave < 8) {
    const int m = lane & 15;          // point row within tile / B column
    const int p = blockBase + wave*16 + m;
    float x = 0.0f, y = 0.0f, z = 0.0f;
    if (p < N) { x = inputs[3*p]; y = inputs[3*p+1]; z = inputs[3*p+2]; }
    // A 16x4 f32: lanes 0-15 -> K0,K1 ; lanes 16-31 -> K2,K3
    v2f a, b;
    a.x = (lane < 16) ? x : z;
    a.y = (lane < 16) ? y : 1.0f;
    // B 4x16 f32: lanes 0-15 -> rows K0,K1 ; lanes 16-31 -> rows K2,K3
    b.x = (lane < 16) ? Ms[0*16 + m] : Ms[2*16 + m];
    b.y = (lane < 16) ? Ms[1*16 + m] : Ms[3*16 + m];
    v8f acc = {0.f,0.f,0.f,0.f,0.f,0.f,0.f,0.f};
    acc = __builtin_amdgcn_wmma_f32_16x16x4_f32(false, a, false, b,
                                                (short)0, acc, false, false);
    // D: vgpr r holds M=r (lanes 0-15) / M=r+8 (lanes 16-31), N = lane&15
#pragma unroll
    for (int r = 0; r < 8; ++r) {
      int pm = (lane < 16) ? r : (r + 8);
      if (m < 9) proj[(wave*16 + pm)*16 + m] = acc[r];
    }
  }
#else
  if (tid < PTS_PER_BLOCK) {
    int p = blockBase + tid;
    float A[4] = {0.f, 0.f, 0.f, 1.f};
    if (p < N) { A[0]=inputs[3*p]; A[1]=inputs[3*p+1]; A[2]=inputs[3*p+2]; }
#pragma unroll
    for (int col = 0; col < 9; ++col)
      proj[tid*16+col] = A[0]*Ms[col] + A[1]*Ms[16+col] + A[2]*Ms[32+col] + Ms[48+col];
  }
#endif
  __syncthreads();

  // ---- Phase 3: per-point projection -> clipped (h,w) -> per-scale offsets ----
  if (tid < PTS_PER_BLOCK) {
    int p = blockBase + tid;
    if (p < N) {
      size_t row = (size_t)p * 2883;
      out[row+0] = inputs[3*p]; out[row+1] = inputs[3*p+1]; out[row+2] = inputs[3*p+2];
      const int   sizes[4]  = {56, 28, 14, 7};
      const float iratio[4] = {0.25f, 0.125f, 0.0625f, 0.03125f}; // / {4,8,16,32}
#pragma unroll
      for (int v = 0; v < 3; ++v) {
        float X = proj[tid*16 + 3*v + 0];
        float Y = proj[tid*16 + 3*v + 1];
        float Z = proj[tid*16 + 3*v + 2];
        float negZ = -Z;
        float h = 248.0f * (-Y) / negZ + 112.0f;
        float w = 248.0f * X / negZ + 112.0f;
        if (h != h) h = 0.0f;                 // nan_to_num; +/-inf handled by clamps
        if (w != w) w = 0.0f;
        h = fminf(fmaxf(h, 0.0f), 223.0f);
        w = fminf(fmaxf(w, 0.0f), 223.0f);
#pragma unroll
        for (int s = 0; s < 4; ++s) {
          int ih = (int)(h * iratio[s]);
          int iw = (int)(w * iratio[s]);
          offs[tid*12 + v*4 + s] = ih * sizes[s] + iw;
        }
      }
    }
  }
  if (wave == 0) wait_tdm_done();  // TDM issuer waits; barrier publishes LDS data
  __syncthreads();

  // ---- Phase 4: gather + max/mean/std streaming. One wave per point. ----
#define EMIT3(jj, va, vb, vc)                                                  \
  do {                                                                         \
    float mx = fmaxf((va), fmaxf((vb), (vc)));                                 \
    float mn = ((va) + (vb) + (vc)) * (1.0f / 3.0f);                           \
    float da = (va) - mn, db = (vb) - mn, dc = (vc) - mn;                      \
    float sd = FAST_SQRT((da*da + db*db + dc*dc) * (1.0f / 3.0f));             \
    __builtin_nontemporal_store(mx, out + row + 3    + (jj));                  \
    __builtin_nontemporal_store(mn, out + row + 963  + (jj));                  \
    __builtin_nontemporal_store(sd, out + row + 1923 + (jj));                  \
  } while (0)

  for (int g = 0; g < PTS_PER_BLOCK / 16; ++g) {
    const int lp = g*16 + wave;
    const int p  = blockBase + lp;
    if (p >= N) continue;
    int off[12];
#pragma unroll
    for (int i = 0; i < 12; ++i) off[i] = offs[lp*12 + i];
    const size_t row = (size_t)p * 2883;

    // level 0: channels 0..63, 56x56 planes from L2
    {
      const float* b0 = feat0 + off[0];
      const float* b1 = feat0 + off[4];
      const float* b2 = feat0 + off[8];
#pragma unroll
      for (int it = 0; it < 2; ++it) {
        int j = it*32 + lane;
        size_t co = (size_t)j * 3136;
        EMIT3(j, b0[co], b1[co], b2[co]);
      }
    }
    // level 1: channels 64..191, 28x28 planes from L2
    {
      const float* b0 = feat1 + off[1];
      const float* b1 = feat1 + off[5];
      const float* b2 = feat1 + off[9];
#pragma unroll
      for (int it = 0; it < 4; ++it) {
        int j = 64 + it*32 + lane;
        size_t co = (size_t)(j - 64) * 784;
        EMIT3(j, b0[co], b1[co], b2[co]);
      }
    }
    // level 2: channels 192..447, 14x14 planes from LDS
    {
      const float* b0 = f2s + off[2];
      const float* b1 = f2s + off[6];
      const float* b2 = f2s + off[10];
#pragma unroll
      for (int it = 0; it < 8; ++it) {
        int j = 192 + it*32 + lane;
        int co = (j - 192) * 196;
        EMIT3(j, b0[co], b1[co], b2[co]);
      }
    }
    // level 3: channels 448..959, 7x7 planes from LDS
    {
      const float* b0 = f3s + off[3];
      const float* b1 = f3s + off[7];
      const float* b2 = f3s + off[11];
#pragma unroll
      for (int it = 0; it < 16; ++it) {
        int j = 448 + it*32 + lane;
        int co = (j - 448) * 49;
        EMIT3(j, b0[co], b1[co], b2[co]);
      }
    }
  }
#undef EMIT3
}

extern "C" void kernel_launch(void* const* d_in, const int* in_sizes, int n_in,
                              void* d_out, int out_size, void* d_ws, size_t ws_size,
                              hipStream_t stream) {
  const float* inputs  = (const float*)d_in[0];
  const float* cameras = (const float*)d_in[1];
  const float* feat0   = (const float*)d_in[2];
  const float* feat1   = (const float*)d_in[3];
  const float* feat2   = (const float*)d_in[4];
  const float* feat3   = (const float*)d_in[5];
  float* out = (float*)d_out;
  const int N = in_sizes[0] / 3;
  const int blocks = (N + PTS_PER_BLOCK - 1) / PTS_PER_BLOCK;
  static bool attr_set = false;  // host-side config only; same value every call
  if (!attr_set) {
    (void)hipFuncSetAttribute((const void*)graphproj_kernel,
                              hipFuncAttributeMaxDynamicSharedMemorySize, LDS_BYTES);
    attr_set = true;
  }
  graphproj_kernel<<<blocks, THREADS, LDS_BYTES, stream>>>(
      inputs, cameras, feat0, feat1, feat2, feat3, out, N);
}